// NonLocalBlock1D_46626164965557
// MI455X (gfx1250) — compile-verified
//
#include <hip/hip_runtime.h>

// Problem constants (from setup_inputs): B=4, C=256, Ci=128, T=8192, N=T/2
#define Bk  4
#define Cc  256
#define Cin 128
#define Tt  8192
#define Npool 4096
#define BN_EPS 1e-5f

typedef float v2f __attribute__((ext_vector_type(2)));
typedef float v8f __attribute__((ext_vector_type(8)));

// V_WMMA_F32_16X16X4_F32 : D(16x16 f32) = A(16x4 f32) * B(4x16 f32) + C
// 8-arg pattern: (neg_a, A, neg_b, B, c_mod, C, reuse_a, reuse_b)
__device__ __forceinline__ v8f wmma4(v2f a, v2f b, v8f c) {
  return __builtin_amdgcn_wmma_f32_16x16x4_f32(false, a, false, b, (short)0, c,
                                               false, false);
}

// ---------------------------------------------------------------------------
// Kernel 1: 1x1 conv as GEMM  out[b,o,t] = sum_c w[o,c] x[b,c,t] + bias[o]
// POOL=1: each wave covers 32 t's and writes maxpooled 16 n's.
// One wave per 16(o) x 16/32(t) tile.
// ---------------------------------------------------------------------------
template <int POOL>
__global__ __launch_bounds__(256) void conv1x1_kernel(
    const float* __restrict__ x, const float* __restrict__ w,
    const float* __restrict__ bias, float* __restrict__ out) {
  const int TW = POOL ? 32 : 16;
  const int tTiles = Tt / TW;
  int wid  = (blockIdx.x * blockDim.x + threadIdx.x) >> 5;
  int lane = threadIdx.x & 31;
  int half = lane >> 4, lm = lane & 15;
  int tt = wid % tTiles;
  int rt = (wid / tTiles) & 7;          // Cin/16 = 8 row tiles
  int b  = wid / (tTiles * 8);
  int o0 = rt * 16, t0 = tt * TW;

  const float* xb = x + (size_t)b * Cc * Tt;
  v8f acc0 = {};
  v8f acc1 = {};
  for (int ks = 0; ks < Cc / 4; ++ks) {
    int k = ks * 4 + half * 2;
    v2f a = *(const v2f*)(w + (size_t)(o0 + lm) * Cc + k);     // A[row=o, k]
    v2f b0;
    b0.x = xb[(size_t)k * Tt + t0 + lm];                        // B[k, col=t]
    b0.y = xb[(size_t)(k + 1) * Tt + t0 + lm];
    acc0 = wmma4(a, b0, acc0);
    if (POOL) {
      v2f b1;
      b1.x = xb[(size_t)k * Tt + t0 + 16 + lm];
      b1.y = xb[(size_t)(k + 1) * Tt + t0 + 16 + lm];
      acc1 = wmma4(a, b1, acc1);
    }
  }

  if (!POOL) {
    float* ob = out + ((size_t)b * Cin + o0) * Tt + t0;
    for (int r = 0; r < 8; ++r) {
      int row = r + half * 8;
      ob[(size_t)row * Tt + lm] = acc0[r] + bias[o0 + row];
    }
  } else {
    // maxpool2 across adjacent t lanes, then add bias (bias commutes with max)
    float* ob = out + ((size_t)b * Cin + o0) * Npool + (t0 >> 1);
    for (int r = 0; r < 8; ++r) {
      int row = r + half * 8;
      float v0 = acc0[r], v1 = acc1[r];
      float m0 = fmaxf(v0, __shfl_xor(v0, 1));
      float m1 = fmaxf(v1, __shfl_xor(v1, 1));
      if ((lm & 1) == 0) {
        float bz = bias[o0 + row];
        ob[(size_t)row * Npool + (lm >> 1)]     = m0 + bz;
        ob[(size_t)row * Npool + 8 + (lm >> 1)] = m1 + bz;
      }
    }
  }
}

// ---------------------------------------------------------------------------
// Kernel 2: M[b,cp,c] = (1/N) * sum_n phi[b,cp,n] * g[b,c,n]   (128x128, K=4096)
// One wave per 16x16 tile, full K. Both fragments are contiguous float2 loads.
// ---------------------------------------------------------------------------
__global__ __launch_bounds__(256) void mmat_kernel(
    const float* __restrict__ phi, const float* __restrict__ g,
    float* __restrict__ Mout) {
  int wid  = (blockIdx.x * blockDim.x + threadIdx.x) >> 5;
  int lane = threadIdx.x & 31;
  int half = lane >> 4, lm = lane & 15;
  int ct  = wid & 7;
  int cpt = (wid >> 3) & 7;
  int b   = wid >> 6;
  const float* pb = phi + ((size_t)b * Cin + cpt * 16) * Npool;
  const float* gb = g   + ((size_t)b * Cin + ct * 16) * Npool;
  v8f acc = {};
  for (int ks = 0; ks < Npool / 4; ++ks) {
    int k = ks * 4 + half * 2;
    v2f a  = *(const v2f*)(pb + (size_t)lm * Npool + k);  // A[row=cp, k=n]
    v2f bb = *(const v2f*)(gb + (size_t)lm * Npool + k);  // B[k=n, col=c]
    acc = wmma4(a, bb, acc);
  }
  const float scale = 1.0f / (float)Npool;
  float* Mb = Mout + (size_t)b * Cin * Cin + (size_t)(cpt * 16) * Cin + ct * 16;
  for (int r = 0; r < 8; ++r) {
    int row = r + half * 8;
    Mb[(size_t)row * Cin + lm] = acc[r] * scale;
  }
}

// ---------------------------------------------------------------------------
// Kernel 3: per (b, t-tile of 16):  y = M^T * theta  (128x16, in LDS),
//           then wy = w_w * y + w_b (256x16) written to d_out.
// 8 waves: phase1 one y row-tile each; phase2 two wy row-tiles each.
// ---------------------------------------------------------------------------
__global__ __launch_bounds__(256) void y_wy_kernel(
    const float* __restrict__ theta, const float* __restrict__ Mmat,
    const float* __restrict__ ww, const float* __restrict__ wb,
    float* __restrict__ outp) {
  __shared__ float ylds[128 * 17];  // pitch 17 to dodge bank conflicts
  int wg = blockIdx.x;              // Bk * (Tt/16) = 2048
  int b  = wg >> 9;
  int t0 = (wg & 511) * 16;
  int wv   = threadIdx.x >> 5;      // 0..7
  int lane = threadIdx.x & 31;
  int half = lane >> 4, lm = lane & 15;

  const float* thb = theta + (size_t)b * Cin * Tt;
  const float* Mb  = Mmat + (size_t)b * Cin * Cin;

  // phase 1: y[c0..c0+15, t0..t0+15] = (M^T theta) tile
  int c0 = wv * 16;
  v8f acc = {};
  for (int ks = 0; ks < Cin / 4; ++ks) {
    int k = ks * 4 + half * 2;
    v2f a;                                   // A[row=c, k=cp] = M[cp, c]
    a.x = Mb[(size_t)k * Cin + c0 + lm];
    a.y = Mb[(size_t)(k + 1) * Cin + c0 + lm];
    v2f bb;                                  // B[k=cp, col=t] = theta[cp, t]
    bb.x = thb[(size_t)k * Tt + t0 + lm];
    bb.y = thb[(size_t)(k + 1) * Tt + t0 + lm];
    acc = wmma4(a, bb, acc);
  }
  for (int r = 0; r < 8; ++r) {
    int row = c0 + r + half * 8;
    ylds[row * 17 + lm] = acc[r];
  }
  __syncthreads();

  // phase 2: wy tiles (o0 = wv*16 and (wv+8)*16)
  for (int ht = 0; ht < 2; ++ht) {
    int o0 = (wv + ht * 8) * 16;
    v8f a2 = {};
    for (int ks = 0; ks < Cin / 4; ++ks) {
      int k = ks * 4 + half * 2;
      v2f a = *(const v2f*)(ww + (size_t)(o0 + lm) * Cin + k);  // A[o, ci]
      v2f bb;                                                    // B[ci, t]
      bb.x = ylds[k * 17 + lm];
      bb.y = ylds[(k + 1) * 17 + lm];
      a2 = wmma4(a, bb, a2);
    }
    float* ob = outp + ((size_t)b * Cc + o0) * Tt + t0;
    for (int r = 0; r < 8; ++r) {
      int row = r + half * 8;
      ob[(size_t)row * Tt + lm] = a2[r] + wb[o0 + row];
    }
  }
}

// ---------------------------------------------------------------------------
// Kernel 4: per-channel sums over (B,T) of wy (deterministic tree reduction)
// ---------------------------------------------------------------------------
__global__ __launch_bounds__(256) void bn_stats_kernel(
    const float* __restrict__ wy, float* __restrict__ sumb,
    float* __restrict__ sumsqb) {
  __shared__ float ls[256], ls2[256];
  int c = blockIdx.x, tid = threadIdx.x;
  float s = 0.f, s2 = 0.f;
  for (int b = 0; b < Bk; ++b) {
    const float* p = wy + ((size_t)b * Cc + c) * Tt;
    for (int t = tid; t < Tt; t += 256) {
      float v = p[t];
      s += v; s2 += v * v;
    }
  }
  ls[tid] = s; ls2[tid] = s2;
  __syncthreads();
  for (int off = 128; off > 0; off >>= 1) {
    if (tid < off) { ls[tid] += ls[tid + off]; ls2[tid] += ls2[tid + off]; }
    __syncthreads();
  }
  if (tid == 0) { sumb[c] = ls[0]; sumsqb[c] = ls2[0]; }
}

__global__ void bn_finalize_kernel(const float* __restrict__ sumb,
                                   const float* __restrict__ sumsqb,
                                   const float* __restrict__ gamma,
                                   const float* __restrict__ beta,
                                   float* __restrict__ scaleb,
                                   float* __restrict__ shiftb) {
  int c = threadIdx.x;  // 256 threads, 1 block
  float inv  = 1.0f / (float)(Bk * Tt);
  float mean = sumb[c] * inv;
  float var  = sumsqb[c] * inv - mean * mean;
  float sc   = gamma[c] * rsqrtf(var + BN_EPS);
  scaleb[c]  = sc;
  shiftb[c]  = beta[c] - mean * sc;
}

// out = scale[c]*wy + shift[c] + x   (wy already resident in d_out)
__global__ __launch_bounds__(256) void bn_apply_kernel(
    const float* __restrict__ x, const float* __restrict__ scaleb,
    const float* __restrict__ shiftb, float* outp) {
  size_t total4 = (size_t)Bk * Cc * Tt / 4;
  size_t stride = (size_t)gridDim.x * blockDim.x;
  for (size_t i = (size_t)blockIdx.x * blockDim.x + threadIdx.x; i < total4;
       i += stride) {
    int c = (int)(((i * 4) / Tt) % Cc);
    float4 wy = ((const float4*)outp)[i];
    float4 xv = ((const float4*)x)[i];
    float sc = scaleb[c], sh = shiftb[c];
    float4 r;
    r.x = sc * wy.x + sh + xv.x;
    r.y = sc * wy.y + sh + xv.y;
    r.z = sc * wy.z + sh + xv.z;
    r.w = sc * wy.w + sh + xv.w;
    ((float4*)outp)[i] = r;
  }
}

// ---------------------------------------------------------------------------
extern "C" void kernel_launch(void* const* d_in, const int* in_sizes, int n_in,
                              void* d_out, int out_size, void* d_ws,
                              size_t ws_size, hipStream_t stream) {
  (void)in_sizes; (void)n_in; (void)out_size; (void)ws_size;
  const float* x       = (const float*)d_in[0];
  const float* theta_w = (const float*)d_in[1];
  const float* theta_b = (const float*)d_in[2];
  const float* phi_w   = (const float*)d_in[3];
  const float* phi_b   = (const float*)d_in[4];
  const float* g_w     = (const float*)d_in[5];
  const float* g_b     = (const float*)d_in[6];
  const float* w_w     = (const float*)d_in[7];
  const float* w_b     = (const float*)d_in[8];
  const float* gamma   = (const float*)d_in[9];
  const float* beta    = (const float*)d_in[10];
  float* outp = (float*)d_out;

  // workspace layout (floats)
  float* ws    = (float*)d_ws;
  float* theta = ws;                                   // 4*128*8192 = 4194304
  float* phiP  = theta + (size_t)Bk * Cin * Tt;        // 2097152
  float* gP    = phiP + (size_t)Bk * Cin * Npool;      // 2097152
  float* Mbuf  = gP + (size_t)Bk * Cin * Npool;        // 65536
  float* sumb  = Mbuf + (size_t)Bk * Cin * Cin;        // 256
  float* sumsq = sumb + Cc;                            // 256
  float* scl   = sumsq + Cc;                           // 256
  float* shf   = scl + Cc;                             // 256  (~32.3 MB total)

  // 1) theta = conv(theta_w) ; phiP/gP = maxpool2(conv(...))
  conv1x1_kernel<0><<<2048, 256, 0, stream>>>(x, theta_w, theta_b, theta);
  conv1x1_kernel<1><<<1024, 256, 0, stream>>>(x, phi_w, phi_b, phiP);
  conv1x1_kernel<1><<<1024, 256, 0, stream>>>(x, g_w, g_b, gP);

  // 2) M = phiP * gP^T / N   (256 waves)
  mmat_kernel<<<32, 256, 0, stream>>>(phiP, gP, Mbuf);

  // 3) y = M^T theta ; wy = w_w*y + w_b  -> d_out
  y_wy_kernel<<<2048, 256, 0, stream>>>(theta, Mbuf, w_w, w_b, outp);

  // 4) batchnorm statistics + finalize + apply (+ residual)
  bn_stats_kernel<<<Cc, 256, 0, stream>>>(outp, sumb, sumsq);
  bn_finalize_kernel<<<1, 256, 0, stream>>>(sumb, sumsq, gamma, beta, scl, shf);
  bn_apply_kernel<<<2048, 256, 0, stream>>>(x, scl, shf, outp);
}